// MulLayer_67327907332267
// MI455X (gfx1250) — compile-verified
//
#include <hip/hip_runtime.h>

typedef __attribute__((ext_vector_type(2))) float v2f;
typedef __attribute__((ext_vector_type(8))) float v8f;

#define HH 256
#define WW 256
#define HWN 65536
#define CINC 256
#define KMASK 9
#define MINPIX 10
#define COVCHUNK 2048

__device__ __forceinline__ v8f wmma4(v2f a, v2f b, v8f c) {
  // D = A(16x4, f32) * B(4x16, f32) + C(16x16, f32)
  return __builtin_amdgcn_wmma_f32_16x16x4_f32(false, a, false, b, (short)0, c,
                                               false, false);
}

// ---------------- small helpers ----------------

__global__ __launch_bounds__(256) void zero_kernel(unsigned* __restrict__ p, int n) {
  int i = blockIdx.x * 256 + threadIdx.x;
  if (i < n) p[i] = 0u;
}

// Pack 9 masks into per-pixel bitfield; accumulate per-mask pixel counts.
__global__ __launch_bounds__(256) void mbits_kernel(const int* __restrict__ masks,
                                                    unsigned* __restrict__ mbits,
                                                    int* __restrict__ cnt) {
  __shared__ int lc[KMASK];
  int tid = threadIdx.x;
  if (tid < KMASK) lc[tid] = 0;
  __syncthreads();
  int p = blockIdx.x * 256 + tid;
  unsigned b = 0;
#pragma unroll
  for (int i = 0; i < KMASK; ++i) {
    if (masks[(size_t)i * HWN + p] == 1) {
      b |= 1u << i;
      atomicAdd(&lc[i], 1);
    }
  }
  mbits[p] = b;
  __syncthreads();
  if (tid < KMASK) atomicAdd(&cnt[tid], lc[tid]);
}

// act[0]=bits(scnt>=10), act[1]=bits(ccnt>=10), act[2]=both
__global__ void act_kernel(const int* __restrict__ ccnt, const int* __restrict__ scnt,
                           unsigned* __restrict__ act) {
  if (threadIdx.x == 0 && blockIdx.x == 0) {
    unsigned as = 0, ac = 0;
    for (int i = 0; i < KMASK; ++i) {
      if (scnt[i] >= MINPIX) as |= 1u << i;
      if (ccnt[i] >= MINPIX) ac |= 1u << i;
    }
    act[0] = as; act[1] = ac; act[2] = as & ac;
  }
}

// mean[i][c] = sum_{p in mask i} x[c,p] / max(n_i,1).  One block per channel c.
__global__ __launch_bounds__(256) void masked_mean_kernel(
    const float* __restrict__ x, const unsigned* __restrict__ mbits,
    const int* __restrict__ cnt, float* __restrict__ mean) {
  int c = blockIdx.x;
  int tid = threadIdx.x;
  const float* xc = x + (size_t)c * HWN;
  float s[KMASK];
#pragma unroll
  for (int i = 0; i < KMASK; ++i) s[i] = 0.f;
  for (int p = tid; p < HWN; p += 256) {
    float v = xc[p];
    unsigned b = mbits[p];
#pragma unroll
    for (int i = 0; i < KMASK; ++i)
      if ((b >> i) & 1) s[i] += v;
  }
  __shared__ float red[256];
  for (int i = 0; i < KMASK; ++i) {
    red[tid] = s[i];
    __syncthreads();
    for (int o = 128; o > 0; o >>= 1) {
      if (tid < o) red[tid] += red[tid + o];
      __syncthreads();
    }
    if (tid == 0) {
      int n = cnt[i];
      mean[(size_t)i * CINC + c] = red[0] / (float)(n > 1 ? n : 1);
    }
    __syncthreads();
  }
}

// fsm[c,p] = x[c,p] - mean[idx][c] where idx = highest active mask containing p
__global__ __launch_bounds__(256) void fsm_kernel(
    const float* __restrict__ x, const unsigned* __restrict__ mbits,
    const unsigned* __restrict__ actp, const float* __restrict__ mean,
    float* __restrict__ out) {
  size_t t = (size_t)blockIdx.x * 256 + threadIdx.x;
  int p = (int)(t & (HWN - 1));
  int c = (int)(t >> 16);
  unsigned bits = mbits[p] & actp[0];
  float v = x[t];
  if (bits) v -= mean[(size_t)(31 - __clz((int)bits)) * CINC + c];
  out[t] = v;
}

// ---------------- conv as implicit GEMM on f32 WMMA ----------------
// Wave computes 32 output channels x 64 pixels (2 cout tiles x 4 pixel tiles,
// 8 accumulators) so each B (activation) load feeds two WMMAs.
// out[Cout,HW], in[Cin,HW], wgt[Cout,Cin,ksz,ksz] (OIHW).
// grid = (HW/64, Cout/32), block = 32.
__global__ __launch_bounds__(32) void conv_wmma_kernel(
    const float* __restrict__ in, const float* __restrict__ wgt,
    const float* __restrict__ bias, float* __restrict__ out,
    int Cin, int ksz, int pad, int relu) {
  const int lane = threadIdx.x;
  const int half = lane >> 4;
  const int col = lane & 15;
  const int p0 = blockIdx.x * 64;
  const int co0 = blockIdx.y * 32;
  const int y0 = p0 >> 8;        // W == 256
  const int x0 = p0 & (WW - 1);
  const int kk = ksz * ksz;
  v8f z = {0.f,0.f,0.f,0.f,0.f,0.f,0.f,0.f};
  v8f a0t0 = z, a0t1 = z, a0t2 = z, a0t3 = z;   // cout tile 0
  v8f a1t0 = z, a1t1 = z, a1t2 = z, a1t3 = z;   // cout tile 1
  const float* wrow0 = wgt + (size_t)(co0 + col) * Cin * kk;       // A0 row
  const float* wrow1 = wgt + (size_t)(co0 + 16 + col) * Cin * kk;  // A1 row
  for (int r = 0; r < ksz; ++r) {
    int yy = y0 + r - pad;
    if (yy < 0 || yy >= HH) continue;
    for (int s = 0; s < ksz; ++s) {
      int xb = x0 + col + s - pad;
      // clamp+mask per pixel tile (branchless padding)
      int xc0 = xb,      xl0 = min(max(xc0, 0), WW - 1), d0 = xl0 - xc0;
      int xc1 = xb + 16, xl1 = min(max(xc1, 0), WW - 1), d1 = xl1 - xc1;
      int xc2 = xb + 32, xl2 = min(max(xc2, 0), WW - 1), d2 = xl2 - xc2;
      int xc3 = xb + 48, xl3 = min(max(xc3, 0), WW - 1), d3 = xl3 - xc3;
      float m0 = (d0 == 0) ? 1.f : 0.f;
      float m1 = (d1 == 0) ? 1.f : 0.f;
      float m2 = (d2 == 0) ? 1.f : 0.f;
      float m3 = (d3 == 0) ? 1.f : 0.f;
      const float* ib = in + (size_t)yy * WW + xb;
      const float* wb0 = wrow0 + r * ksz + s;
      const float* wb1 = wrow1 + r * ksz + s;
      for (int ci = 0; ci < Cin; ci += 4) {
        int c2 = ci + 2 * half;          // this lane's K pair
        v2f aA, aB;
        aA.x = wb0[(size_t)c2 * kk];
        aA.y = wb0[(size_t)(c2 + 1) * kk];
        aB.x = wb1[(size_t)c2 * kk];
        aB.y = wb1[(size_t)(c2 + 1) * kk];
        const float* i0 = ib + (size_t)c2 * HWN;
        const float* i1 = i0 + HWN;
        v2f b0, b1, b2, b3;
        b0.x = m0 * i0[d0];      b0.y = m0 * i1[d0];
        b1.x = m1 * i0[16 + d1]; b1.y = m1 * i1[16 + d1];
        b2.x = m2 * i0[32 + d2]; b2.y = m2 * i1[32 + d2];
        b3.x = m3 * i0[48 + d3]; b3.y = m3 * i1[48 + d3];
        a0t0 = wmma4(aA, b0, a0t0);
        a0t1 = wmma4(aA, b1, a0t1);
        a0t2 = wmma4(aA, b2, a0t2);
        a0t3 = wmma4(aA, b3, a0t3);
        a1t0 = wmma4(aB, b0, a1t0);
        a1t1 = wmma4(aB, b1, a1t1);
        a1t2 = wmma4(aB, b2, a1t2);
        a1t3 = wmma4(aB, b3, a1t3);
      }
    }
  }
#pragma unroll
  for (int e = 0; e < 8; ++e) {
    int cr0 = co0 + e + 8 * half;   // D row mapping, cout tile 0
    int cr1 = cr0 + 16;             // cout tile 1
    float bv0 = bias[cr0];
    float bv1 = bias[cr1];
    float v00 = a0t0[e] + bv0, v01 = a0t1[e] + bv0;
    float v02 = a0t2[e] + bv0, v03 = a0t3[e] + bv0;
    float v10 = a1t0[e] + bv1, v11 = a1t1[e] + bv1;
    float v12 = a1t2[e] + bv1, v13 = a1t3[e] + bv1;
    if (relu) {
      v00 = fmaxf(v00, 0.f); v01 = fmaxf(v01, 0.f);
      v02 = fmaxf(v02, 0.f); v03 = fmaxf(v03, 0.f);
      v10 = fmaxf(v10, 0.f); v11 = fmaxf(v11, 0.f);
      v12 = fmaxf(v12, 0.f); v13 = fmaxf(v13, 0.f);
    }
    float* ob0 = out + (size_t)cr0 * HWN + p0 + col;
    float* ob1 = out + (size_t)cr1 * HWN + p0 + col;
    ob0[0] = v00; ob0[16] = v01; ob0[32] = v02; ob0[48] = v03;
    ob1[0] = v10; ob1[16] = v11; ob1[32] = v12; ob1[48] = v13;
  }
}

// ---------------- masked covariance via WMMA ----------------
// covsum[i] += (ff*m_i) @ ff^T over a chunk of pixels. grid=(HW/COVCHUNK, 9).
__global__ __launch_bounds__(32) void cov_wmma_kernel(
    const float* __restrict__ ff, const unsigned* __restrict__ mbits,
    float* __restrict__ covsum) {
  const int lane = threadIdx.x;
  const int half = lane >> 4;
  const int col = lane & 15;
  const int im = blockIdx.y;
  const int p0 = blockIdx.x * COVCHUNK;
  const float* f0 = ff + (size_t)col * HWN;         // rows 0..15
  const float* f1 = ff + (size_t)(16 + col) * HWN;  // rows 16..31
  v8f a00 = {0.f,0.f,0.f,0.f,0.f,0.f,0.f,0.f};
  v8f a01 = a00, a10 = a00, a11 = a00;
  for (int p = p0; p < p0 + COVCHUNK; p += 4) {
    int pk = p + 2 * half;
    v2f u0 = *(const v2f*)(f0 + pk);
    v2f u1 = *(const v2f*)(f1 + pk);
    uint2 mb = *(const uint2*)(mbits + pk);
    float m0 = ((mb.x >> im) & 1) ? 1.f : 0.f;
    float m1 = ((mb.y >> im) & 1) ? 1.f : 0.f;
    v2f ma0; ma0.x = u0.x * m0; ma0.y = u0.y * m1;
    v2f ma1; ma1.x = u1.x * m0; ma1.y = u1.y * m1;
    a00 = wmma4(ma0, u0, a00);
    a01 = wmma4(ma0, u1, a01);
    a10 = wmma4(ma1, u0, a10);
    a11 = wmma4(ma1, u1, a11);
  }
  float* cv = covsum + (size_t)im * 1024;
#pragma unroll
  for (int e = 0; e < 8; ++e) {
    int r = e + 8 * half;
    atomicAdd(&cv[r * 32 + col],             a00[e]);
    atomicAdd(&cv[r * 32 + 16 + col],        a01[e]);
    atomicAdd(&cv[(16 + r) * 32 + col],      a10[e]);
    atomicAdd(&cv[(16 + r) * 32 + 16 + col], a11[e]);
  }
}

// tms[i] = fc_w @ (cov_i / max(n,1)) + fc_b.  grid=(4, 9), block=256.
__global__ __launch_bounds__(256) void tm_kernel(
    const float* __restrict__ fcw, const float* __restrict__ fcb,
    const float* __restrict__ covsum, const int* __restrict__ cnt,
    float* __restrict__ tms) {
  __shared__ float cv[1024];
  int i = blockIdx.y, tid = threadIdx.x;
  int n = cnt[i];
  float inv = 1.f / (float)(n > 1 ? n : 1);
  for (int k = tid; k < 1024; k += 256) cv[k] = covsum[(size_t)i * 1024 + k] * inv;
  __syncthreads();
  int o = blockIdx.x * 256 + tid;
  const float* wr = fcw + (size_t)o * 1024;
  float s = fcb[o];
  for (int k = 0; k < 1024; ++k) s += wr[k] * cv[k];
  tms[(size_t)i * 1024 + o] = s;
}

// tmf[i] = sMs[i] @ cMs[i]  (32x32 each). grid=9, block=1024.
__global__ __launch_bounds__(1024) void tmprod_kernel(
    const float* __restrict__ sMs, const float* __restrict__ cMs,
    float* __restrict__ tmf) {
  int i = blockIdx.x, tid = threadIdx.x;
  int r = tid >> 5, c = tid & 31;
  const float* A = sMs + (size_t)i * 1024;
  const float* B = cMs + (size_t)i * 1024;
  float s = 0.f;
#pragma unroll
  for (int k = 0; k < 32; ++k) s += A[r * 32 + k] * B[k * 32 + c];
  tmf[(size_t)i * 1024 + tid] = s;
}

// Final: per-pixel transform select + unzip + style-mean add.
__global__ __launch_bounds__(256) void final_kernel(
    const float* __restrict__ ccf, const float* __restrict__ tmf,
    const unsigned* __restrict__ cmbits, const unsigned* __restrict__ actp,
    const float* __restrict__ smean, const float* __restrict__ uw,
    const float* __restrict__ ub, float* __restrict__ out) {
  __shared__ float s_tm[KMASK * 1024];
  __shared__ float s_sm[KMASK * CINC];
  __shared__ float s_ub[256];
  int tid = threadIdx.x;
  for (int j = tid; j < KMASK * 1024; j += 256) s_tm[j] = tmf[j];
  for (int j = tid; j < KMASK * CINC; j += 256) s_sm[j] = smean[j];
  s_ub[tid] = ub[tid];
  __syncthreads();
  int p = blockIdx.x * 256 + tid;
  unsigned bits = cmbits[p] & actp[0];
  int idx = bits ? (31 - __clz((int)bits)) : -1;
  float cv[32], tf[32];
#pragma unroll
  for (int j = 0; j < 32; ++j) cv[j] = ccf[(size_t)j * HWN + p];
  if (idx >= 0) {
    const float* tmr = s_tm + (size_t)idx * 1024;
#pragma unroll
    for (int r = 0; r < 32; ++r) {
      float s = 0.f;
#pragma unroll
      for (int k = 0; k < 32; ++k) s += tmr[r * 32 + k] * cv[k];
      tf[r] = s;
    }
  } else {
#pragma unroll
    for (int r = 0; r < 32; ++r) tf[r] = cv[r];
  }
  for (int c = 0; c < 256; ++c) {
    const float* wr = uw + (size_t)c * 32;
    float s = s_ub[c];
#pragma unroll
    for (int j = 0; j < 32; ++j) s += wr[j] * tf[j];
    if (idx >= 0) s += s_sm[(size_t)idx * CINC + c];
    out[(size_t)c * HWN + p] = s;
  }
}

// ---------------- host-side orchestration ----------------

extern "C" void kernel_launch(void* const* d_in, const int* in_sizes, int n_in,
                              void* d_out, int out_size, void* d_ws, size_t ws_size,
                              hipStream_t stream) {
  (void)in_sizes; (void)n_in; (void)out_size; (void)ws_size;
  const float* cF = (const float*)d_in[0];
  const float* sF = (const float*)d_in[1];
  const int* cmasks = (const int*)d_in[2];
  const int* smasks = (const int*)d_in[3];
  const float* s_c1_w = (const float*)d_in[4];
  const float* s_c1_b = (const float*)d_in[5];
  const float* s_c2_w = (const float*)d_in[6];
  const float* s_c2_b = (const float*)d_in[7];
  const float* s_c3_w = (const float*)d_in[8];
  const float* s_c3_b = (const float*)d_in[9];
  const float* s_fc_w = (const float*)d_in[10];
  const float* s_fc_b = (const float*)d_in[11];
  const float* c_c1_w = (const float*)d_in[12];
  const float* c_c1_b = (const float*)d_in[13];
  const float* c_c2_w = (const float*)d_in[14];
  const float* c_c2_b = (const float*)d_in[15];
  const float* c_c3_w = (const float*)d_in[16];
  const float* c_c3_b = (const float*)d_in[17];
  const float* c_fc_w = (const float*)d_in[18];
  const float* c_fc_b = (const float*)d_in[19];
  const float* compress_w = (const float*)d_in[20];
  const float* compress_b = (const float*)d_in[21];
  const float* unzip_w = (const float*)d_in[22];
  const float* unzip_b = (const float*)d_in[23];

  char* ws = (char*)d_ws;
  size_t off = 0;
  auto take = [&](size_t bytes) -> char* {
    char* p = ws + off;
    off += (bytes + 255) & ~(size_t)255;
    return p;
  };
  // zeroed region first (counts + cov partial sums)
  int* scnt = (int*)take(KMASK * 4);
  int* ccnt = (int*)take(KMASK * 4);
  float* covS = (float*)take(KMASK * 1024 * 4);
  float* covC = (float*)take(KMASK * 1024 * 4);
  size_t zero_bytes = off;
  unsigned* act = (unsigned*)take(4 * 4);
  unsigned* smbits = (unsigned*)take((size_t)HWN * 4);
  unsigned* cmbits = (unsigned*)take((size_t)HWN * 4);
  float* smean = (float*)take(KMASK * CINC * 4);
  float* cmean = (float*)take(KMASK * CINC * 4);
  float* sMs = (float*)take(KMASK * 1024 * 4);
  float* cMs = (float*)take(KMASK * 1024 * 4);
  float* tmf = (float*)take(KMASK * 1024 * 4);
  float* bufA = (float*)take((size_t)CINC * HWN * 4);  // fsm (64 MB)
  float* bufB = (float*)take((size_t)128 * HWN * 4);   // h1 (32 MB)
  float* bufC = (float*)take((size_t)64 * HWN * 4);    // h2 (16 MB)
  float* bufD = (float*)take((size_t)32 * HWN * 4);    // feat (8 MB, reused s then c)
  float* bufF = (float*)take((size_t)32 * HWN * 4);    // ccf (8 MB)

  int nz = (int)(zero_bytes / 4);
  zero_kernel<<<(nz + 255) / 256, 256, 0, stream>>>((unsigned*)scnt, nz);

  mbits_kernel<<<HWN / 256, 256, 0, stream>>>(cmasks, cmbits, ccnt);
  mbits_kernel<<<HWN / 256, 256, 0, stream>>>(smasks, smbits, scnt);
  act_kernel<<<1, 32, 0, stream>>>(ccnt, scnt, act);
  masked_mean_kernel<<<CINC, 256, 0, stream>>>(sF, smbits, scnt, smean);
  masked_mean_kernel<<<CINC, 256, 0, stream>>>(cF, cmbits, ccnt, cmean);

  // ---- style stream ----
  fsm_kernel<<<(CINC * HWN) / 256, 256, 0, stream>>>(sF, smbits, act + 0, smean, bufA);
  conv_wmma_kernel<<<dim3(HWN / 64, 4), 32, 0, stream>>>(bufA, s_c1_w, s_c1_b, bufB, 256, 3, 1, 1);
  conv_wmma_kernel<<<dim3(HWN / 64, 2), 32, 0, stream>>>(bufB, s_c2_w, s_c2_b, bufC, 128, 3, 1, 1);
  conv_wmma_kernel<<<dim3(HWN / 64, 1), 32, 0, stream>>>(bufC, s_c3_w, s_c3_b, bufD, 64, 3, 1, 0);
  cov_wmma_kernel<<<dim3(HWN / COVCHUNK, KMASK), 32, 0, stream>>>(bufD, smbits, covS);
  tm_kernel<<<dim3(4, KMASK), 256, 0, stream>>>(s_fc_w, s_fc_b, covS, scnt, sMs);

  // ---- content stream ----
  fsm_kernel<<<(CINC * HWN) / 256, 256, 0, stream>>>(cF, cmbits, act + 1, cmean, bufA);
  conv_wmma_kernel<<<dim3(HWN / 64, 4), 32, 0, stream>>>(bufA, c_c1_w, c_c1_b, bufB, 256, 3, 1, 1);
  conv_wmma_kernel<<<dim3(HWN / 64, 2), 32, 0, stream>>>(bufB, c_c2_w, c_c2_b, bufC, 128, 3, 1, 1);
  conv_wmma_kernel<<<dim3(HWN / 64, 1), 32, 0, stream>>>(bufC, c_c3_w, c_c3_b, bufD, 64, 3, 1, 0);
  cov_wmma_kernel<<<dim3(HWN / COVCHUNK, KMASK), 32, 0, stream>>>(bufD, cmbits, covC);
  tm_kernel<<<dim3(4, KMASK), 256, 0, stream>>>(c_fc_w, c_fc_b, covC, ccnt, cMs);

  // compress on content fsm (still in bufA): 1x1 conv 256->32
  conv_wmma_kernel<<<dim3(HWN / 64, 1), 32, 0, stream>>>(bufA, compress_w, compress_b, bufF, 256, 1, 0, 0);

  tmprod_kernel<<<KMASK, 1024, 0, stream>>>(sMs, cMs, tmf);

  final_kernel<<<HWN / 256, 256, 0, stream>>>(bufF, tmf, cmbits, act + 2, smean,
                                              unzip_w, unzip_b, (float*)d_out);
}